// Zonotope_17532056502789
// MI455X (gfx1250) — compile-verified
//
#include <hip/hip_runtime.h>
#include <math.h>

typedef __attribute__((ext_vector_type(2))) float v2f;
typedef __attribute__((ext_vector_type(8))) float v8f;

#define N_IN 4096
#define H    4096
#define NOUT 10
#define EPSV 0.01f

// ---------------- Kernel 1: input box ----------------
__global__ void zono_prep(const float* __restrict__ in,
                          float* __restrict__ vals,
                          float* __restrict__ radii) {
  int j = blockIdx.x * blockDim.x + threadIdx.x;
  if (j < N_IN) {
    float u = in[j] + EPSV;
    u = u - fmaxf(u - 1.0f, 0.0f);          // clip upper at 1
    float l = fmaxf(in[j] - EPSV, 0.0f);    // clip lower at 0
    vals[j]  = u + l;                        // torch code: upper+lower (no /2)
    radii[j] = u;                            // values - lower == clipped upper
  }
}

// ---------------- Kernel 2: layer1 matvec + |W1|·radii + zono ReLU ----------------
__global__ __launch_bounds__(256) void zono_layer1(
    const float* __restrict__ W1, const float* __restrict__ b1,
    const float* __restrict__ vals0, const float* __restrict__ radii,
    float* __restrict__ vout1, float* __restrict__ scale1, float* __restrict__ last1) {
  __shared__ float sdv[256], sdr[256];
  int i = blockIdx.x;
  const float* row = W1 + (size_t)i * N_IN;
  float dv = 0.f, dr = 0.f;
  for (int j = threadIdx.x; j < N_IN; j += 256) {
    float w = row[j];
    dv += w * vals0[j];
    dr += fabsf(w) * fabsf(radii[j]);
  }
  sdv[threadIdx.x] = dv; sdr[threadIdx.x] = dr;
  __syncthreads();
  for (int s = 128; s > 0; s >>= 1) {
    if (threadIdx.x < s) {
      sdv[threadIdx.x] += sdv[threadIdx.x + s];
      sdr[threadIdx.x] += sdr[threadIdx.x + s];
    }
    __syncthreads();
  }
  if (threadIdx.x == 0) {
    float v = sdv[0] + b1[i];
    float r = sdr[0];
    float u = v + r, l = v - r;
    float denom = (u - l > 0.f) ? (u - l) : 1.f;
    float slope = u / denom;
    float term  = (1.f - slope) * u * 0.5f;
    bool dead = (u <= 0.f), stable = (l >= 0.f);
    bool crossing = !(dead || stable);
    vout1[i]  = dead ? 0.f : (stable ? v : slope * v + term);
    scale1[i] = dead ? 0.f : (stable ? 1.f : slope);
    last1[i]  = crossing ? term : 0.f;
  }
}

// ---------------- Kernel 3: M2 = (W2 * diag(scale1)) @ W1, fp32 WMMA ----------------
// Pitch 20 floats: (row*20 + k) with k even -> banks for lanes 0..15 are == {0,1} mod 4,
// lanes 16..31 (K offset +2) are == {2,3} mod 4: half-waves disjoint, conflict-free b64
// fragment loads for BOTH A and transposed-B. 20 == 0 mod 4 also keeps float4 staging
// stores of A 16B-aligned.
#define BM 128
#define BN 128
#define BK 16
#define LP 20    // LDS pitch (floats) for As and transposed Bs

__global__ __launch_bounds__(256) void zono_gemm(
    const float* __restrict__ W2, const float* __restrict__ W1,
    const float* __restrict__ scale1, float* __restrict__ M2) {
  __shared__ float As[BM * LP];    // [m][k]
  __shared__ float Bst[BN * LP];   // [n][k]  (transposed tile of W1)
  const int tid  = threadIdx.x;
  const int lane = tid & 31;
  const int wave = tid >> 5;
  const int wm0  = (wave >> 1) * 32;   // 4 wave-rows
  const int wn0  = (wave & 1) * 64;    // 2 wave-cols
  const int bm0  = blockIdx.y * BM;
  const int bn0  = blockIdx.x * BN;
  const int lm   = lane & 15;
  const int lg   = lane >> 4;          // half-wave: owns K pair (lg*2, lg*2+1)

  v8f acc[2][4];
#pragma unroll
  for (int a = 0; a < 2; ++a)
#pragma unroll
    for (int b = 0; b < 4; ++b) acc[a][b] = (v8f)0.f;

  for (int kt = 0; kt < H; kt += BK) {
    // Stage A tile (BMxBK) as float4, fold in scale1 column scaling
#pragma unroll
    for (int it = 0; it < (BM * BK) / (256 * 4); ++it) {
      int idx = it * 256 + tid;          // [0, 512)
      int m  = idx >> 2;                 // 0..127
      int k4 = (idx & 3) * 4;            // 0,4,8,12
      float4 w = *reinterpret_cast<const float4*>(
          W2 + (size_t)(bm0 + m) * H + kt + k4);
      float4 s = *reinterpret_cast<const float4*>(scale1 + kt + k4);
      w.x *= s.x; w.y *= s.y; w.z *= s.z; w.w *= s.w;
      *reinterpret_cast<float4*>(&As[m * LP + k4]) = w;
    }
    // Stage B tile (BKxBN) as float4, store transposed [n][k]
#pragma unroll
    for (int it = 0; it < (BK * BN) / (256 * 4); ++it) {
      int idx = it * 256 + tid;          // [0, 512)
      int k  = idx >> 5;                 // 0..15
      int j4 = (idx & 31) * 4;           // 0..124 step 4
      float4 w = *reinterpret_cast<const float4*>(
          W1 + (size_t)(kt + k) * N_IN + bn0 + j4);
      Bst[(j4 + 0) * LP + k] = w.x;
      Bst[(j4 + 1) * LP + k] = w.y;
      Bst[(j4 + 2) * LP + k] = w.z;
      Bst[(j4 + 3) * LP + k] = w.w;
    }
    __syncthreads();
#pragma unroll
    for (int ks = 0; ks < BK; ks += 4) {
      int kb = ks + lg * 2;  // K pair for this half-wave (VGPR0=K, VGPR1=K+1)
      v2f afr[2], bfr[4];
#pragma unroll
      for (int sm = 0; sm < 2; ++sm) {
        int row = wm0 + sm * 16 + lm;
        afr[sm] = *reinterpret_cast<const v2f*>(&As[row * LP + kb]);
      }
#pragma unroll
      for (int sn = 0; sn < 4; ++sn) {
        int col = wn0 + sn * 16 + lm;
        bfr[sn] = *reinterpret_cast<const v2f*>(&Bst[col * LP + kb]);
      }
#pragma unroll
      for (int sm = 0; sm < 2; ++sm)
#pragma unroll
        for (int sn = 0; sn < 4; ++sn)
          acc[sm][sn] = __builtin_amdgcn_wmma_f32_16x16x4_f32(
              false, afr[sm], false, bfr[sn], (short)0, acc[sm][sn], false, false);
    }
    __syncthreads();
  }
  // Write back: C layout VGPR r -> M = r + 8*halfwave, N = lane%16
#pragma unroll
  for (int sm = 0; sm < 2; ++sm)
#pragma unroll
    for (int sn = 0; sn < 4; ++sn)
#pragma unroll
      for (int r = 0; r < 8; ++r) {
        int m = bm0 + wm0 + sm * 16 + r + lg * 8;
        int n = bn0 + wn0 + sn * 16 + lm;
        M2[(size_t)m * N_IN + n] = acc[sm][sn][r];
      }
}

// ---------------- Kernel 4: v2raw = W2 @ vout1 + b2 ; last2raw = W2 @ last1 ----------------
__global__ __launch_bounds__(256) void zono_vec2(
    const float* __restrict__ W2, const float* __restrict__ b2,
    const float* __restrict__ vout1, const float* __restrict__ last1,
    float* __restrict__ v2raw, float* __restrict__ last2raw) {
  __shared__ float sv[256], sl[256];
  int i = blockIdx.x;
  const float* row = W2 + (size_t)i * H;
  float dv = 0.f, dl = 0.f;
  for (int j = threadIdx.x; j < H; j += 256) {
    float w = row[j];
    dv += w * vout1[j];
    dl += w * last1[j];
  }
  sv[threadIdx.x] = dv; sl[threadIdx.x] = dl;
  __syncthreads();
  for (int s = 128; s > 0; s >>= 1) {
    if (threadIdx.x < s) {
      sv[threadIdx.x] += sv[threadIdx.x + s];
      sl[threadIdx.x] += sl[threadIdx.x + s];
    }
    __syncthreads();
  }
  if (threadIdx.x == 0) {
    v2raw[i]    = sv[0] + b2[i];
    last2raw[i] = sl[0];
  }
}

// ---------------- Kernel 5: ReLU-2 relaxation (needs |M2| row sums) ----------------
__global__ __launch_bounds__(256) void zono_relu2(
    const float* __restrict__ M2, const float* __restrict__ radii,
    const float* __restrict__ v2raw, const float* __restrict__ last2raw,
    float* __restrict__ vout2, float* __restrict__ scale2, float* __restrict__ last2b) {
  __shared__ float sdr[256];
  int i = blockIdx.x;
  const float* row = M2 + (size_t)i * N_IN;
  float dr = 0.f;
  for (int j = threadIdx.x; j < N_IN; j += 256)
    dr += fabsf(row[j]) * fabsf(radii[j]);
  sdr[threadIdx.x] = dr;
  __syncthreads();
  for (int s = 128; s > 0; s >>= 1) {
    if (threadIdx.x < s) sdr[threadIdx.x] += sdr[threadIdx.x + s];
    __syncthreads();
  }
  if (threadIdx.x == 0) {
    float v = v2raw[i];
    float r = sdr[0] + fabsf(last2raw[i]);
    float u = v + r, l = v - r;
    float denom = (u - l > 0.f) ? (u - l) : 1.f;
    float slope = u / denom;
    float term  = (1.f - slope) * u * 0.5f;
    bool dead = (u <= 0.f), stable = (l >= 0.f);
    bool crossing = !(dead || stable);
    vout2[i]  = dead ? 0.f : (stable ? v : slope * v + term);
    scale2[i] = dead ? 0.f : (stable ? 1.f : slope);
    last2b[i] = crossing ? term : 0.f;
  }
}

// ---------------- Kernel 6a: partial G over K-chunks (16x more parallelism) ----------------
// Gpart[kb][i][j] = sum_{k in chunk kb} W3[i,k]*scale2[k]*M2[k,j]
__global__ __launch_bounds__(256) void zono_final_part(
    const float* __restrict__ M2, const float* __restrict__ W3,
    const float* __restrict__ scale2, float* __restrict__ Gpart) {
  __shared__ float w3s[NOUT * 256];
  int tid = threadIdx.x;
  int j = blockIdx.x * 256 + tid;        // output column
  int kbase = blockIdx.y * 256;          // K chunk
  float s2 = scale2[kbase + tid];
#pragma unroll
  for (int i = 0; i < NOUT; ++i)
    w3s[i * 256 + tid] = W3[(size_t)i * H + kbase + tid] * s2;
  __syncthreads();
  float g[NOUT];
#pragma unroll
  for (int i = 0; i < NOUT; ++i) g[i] = 0.f;
  for (int kk = 0; kk < 256; ++kk) {
    float m2 = M2[(size_t)(kbase + kk) * N_IN + j];
#pragma unroll
    for (int i = 0; i < NOUT; ++i) g[i] += w3s[i * 256 + kk] * m2;
  }
#pragma unroll
  for (int i = 0; i < NOUT; ++i)
    Gpart[((size_t)blockIdx.y * NOUT + i) * N_IN + j] = g[i];
}

// ---------------- Kernel 6b: finish G columns, |G|·radii block-reductions ----------------
__global__ __launch_bounds__(256) void zono_final_reduce(
    const float* __restrict__ Gpart, const float* __restrict__ radii,
    int nkb, float* __restrict__ partial) {
  __shared__ float red[256];
  int tid = threadIdx.x;
  int j = blockIdx.x * 256 + tid;
  float rj = fabsf(radii[j]);
  for (int i = 0; i < NOUT; ++i) {
    float g = 0.f;
    for (int kb = 0; kb < nkb; ++kb)     // fixed order -> deterministic
      g += Gpart[((size_t)kb * NOUT + i) * N_IN + j];
    red[tid] = fabsf(g) * rj;
    __syncthreads();
    for (int s = 128; s > 0; s >>= 1) {
      if (tid < s) red[tid] += red[tid + s];
      __syncthreads();
    }
    if (tid == 0) partial[blockIdx.x * NOUT + i] = red[0];
    __syncthreads();
  }
}

// ---------------- Kernel 7: final outputs ----------------
__global__ __launch_bounds__(256) void zono_out(
    const float* __restrict__ W3, const float* __restrict__ b3,
    const float* __restrict__ vout2, const float* __restrict__ scale2,
    const float* __restrict__ last2raw, const float* __restrict__ last2b,
    const float* __restrict__ partial, int nblk, float* __restrict__ out) {
  __shared__ float sv[256], sc1[256], sc2[256];
  int tid = threadIdx.x;
  for (int i = 0; i < NOUT; ++i) {
    float pv = 0.f, p1 = 0.f, p2 = 0.f;
    for (int k = tid; k < H; k += 256) {
      float w = W3[(size_t)i * H + k];
      pv += w * vout2[k];
      p1 += w * scale2[k] * last2raw[k];  // col 4096 of E3
      p2 += w * last2b[k];                // col 4097 of E3
    }
    sv[tid] = pv; sc1[tid] = p1; sc2[tid] = p2;
    __syncthreads();
    for (int s = 128; s > 0; s >>= 1) {
      if (tid < s) { sv[tid] += sv[tid+s]; sc1[tid] += sc1[tid+s]; sc2[tid] += sc2[tid+s]; }
      __syncthreads();
    }
    if (tid == 0) {
      float rb = 0.f;
      for (int b = 0; b < nblk; ++b) rb += partial[b * NOUT + i];  // deterministic order
      float v = sv[0] + b3[i];
      float r = rb + fabsf(sc1[0]) + fabsf(sc2[0]);
      out[i]        = v + r;  // upper
      out[NOUT + i] = v - r;  // lower
    }
    __syncthreads();
  }
}

// ---------------- Host launcher ----------------
extern "C" void kernel_launch(void* const* d_in, const int* in_sizes, int n_in,
                              void* d_out, int out_size, void* d_ws, size_t ws_size,
                              hipStream_t stream) {
  const float* inputs = (const float*)d_in[0];
  const float* W1 = (const float*)d_in[1];
  const float* b1 = (const float*)d_in[2];
  const float* W2 = (const float*)d_in[3];
  const float* b2 = (const float*)d_in[4];
  const float* W3 = (const float*)d_in[5];
  const float* b3 = (const float*)d_in[6];
  float* out = (float*)d_out;
  float* ws  = (float*)d_ws;

  // Workspace layout (floats); ~70 MB total (M2 is 64 MB, Gpart 2.6 MB).
  float* vals0    = ws;
  float* radii    = ws + 4096;
  float* vout1    = ws + 8192;
  float* scale1   = ws + 12288;
  float* last1    = ws + 16384;
  float* v2raw    = ws + 20480;
  float* last2raw = ws + 24576;
  float* vout2    = ws + 28672;
  float* scale2   = ws + 32768;
  float* last2b   = ws + 36864;
  float* partial  = ws + 40960;          // 16 blocks * 10
  float* Gpart    = ws + 49152;          // 16 * 10 * 4096 = 655360
  float* M2       = ws + 720896;         // 4096*4096

  zono_prep<<<(N_IN + 255) / 256, 256, 0, stream>>>(inputs, vals0, radii);
  zono_layer1<<<H, 256, 0, stream>>>(W1, b1, vals0, radii, vout1, scale1, last1);
  zono_vec2<<<H, 256, 0, stream>>>(W2, b2, vout1, last1, v2raw, last2raw);
  zono_gemm<<<dim3(N_IN / BN, H / BM), 256, 0, stream>>>(W2, W1, scale1, M2);
  zono_relu2<<<H, 256, 0, stream>>>(M2, radii, v2raw, last2raw, vout2, scale2, last2b);
  zono_final_part<<<dim3(N_IN / 256, H / 256), 256, 0, stream>>>(M2, W3, scale2, Gpart);
  zono_final_reduce<<<N_IN / 256, 256, 0, stream>>>(Gpart, radii, H / 256, partial);
  zono_out<<<1, 256, 0, stream>>>(W3, b3, vout2, scale2, last2raw, last2b,
                                  partial, N_IN / 256, out);
}